// SignaxOnly_10788957847915
// MI455X (gfx1250) — compile-verified
//
#include <hip/hip_runtime.h>
#include <hip/hip_bf16.h>

// Problem constants (reference: LENGTH=4096, DIM=16, MAX_K=4)
#define NSTEP 4095   // number of path increments
#define K4    4096   // padded K for WMMA (pad row = zeros)
#define DIMS  16
#define L2SZ  256
#define L3SZ  4096
#define KCH   32     // K-chunk staged in LDS per double-buffer slot

typedef __attribute__((ext_vector_type(2))) float v2f;
typedef __attribute__((ext_vector_type(8))) float v8f;
typedef int v4i_vs __attribute__((vector_size(16)));   // 128-bit payload type for async LDS copy

// gfx1250 async global->LDS copy (ASYNCcnt-tracked), if the toolchain exposes it.
#if __has_builtin(__builtin_amdgcn_global_load_async_to_lds_b128) && \
    __has_builtin(__builtin_amdgcn_s_wait_asynccnt)
#define USE_ASYNC_LDS 1
#else
#define USE_ASYNC_LDS 0
#endif

__global__ void k_zero(float* p, int n) {
  int i = blockIdx.x * blockDim.x + threadIdx.x;
  if (i < n) p[i] = 0.0f;
}

// dx[j][d] = x[j+1][d]-x[j][d]  (pad row = 0),  Q[j] = dx⊗dx/2 (256 wide)
__global__ void k_dx_q(const float* __restrict__ x,
                       float* __restrict__ dx, float* __restrict__ Q) {
  int j = blockIdx.x;         // 0..4095
  int t = threadIdx.x;        // 0..255
  __shared__ float sdx[DIMS];
  if (t < DIMS) {
    float d = (j < NSTEP) ? (x[(j + 1) * DIMS + t] - x[j * DIMS + t]) : 0.0f;
    sdx[t] = d;
    dx[j * DIMS + t] = d;
  }
  __syncthreads();
  Q[j * L2SZ + t] = 0.5f * sdx[t >> 4] * sdx[t & 15];
}

// Exclusive prefix sums of dx (width 16). Also emits S1 = total sum.
__global__ void k_scan_c(const float* __restrict__ dx,
                         float* __restrict__ cx, float* __restrict__ out) {
  int d = threadIdx.x;
  if (d >= DIMS) return;
  float acc = 0.0f;
  for (int j = 0; j < K4; ++j) {
    cx[j * DIMS + d] = acc;
    acc += dx[j * DIMS + d];
  }
  out[d] = acc;  // level-1 signature
}

// T[j] = exclusive prefix of level-2 increments (c_j⊗dx_j + Q_j), width 256.
__global__ void k_scan_t(const float* __restrict__ dx, const float* __restrict__ cx,
                         const float* __restrict__ Q, float* __restrict__ T) {
  int col = threadIdx.x;      // (a,b)
  int a = col >> 4, b = col & 15;
  float acc = 0.0f;
  for (int j = 0; j < K4; ++j) {
    T[j * L2SZ + col] = acc;
    acc += cx[j * DIMS + a] * dx[j * DIMS + b] + Q[j * L2SZ + col];
  }
}

// Column-parallel scan of level-3 increments with the U^T D GEMM fused in:
// u = exclusive prefix of  Δ3_j[(a,b,c)] = T_j[ab]·dx_j[c] + (c_j[a] + dx_j[a]/3)·Q_j[bc]
// S4[(abc), n] = Σ_j u_j · dx_j[n]    (first S4 term; written with '=', GEMMs += after)
__global__ void k_scan_u_s4(const float* __restrict__ dx, const float* __restrict__ cx,
                            const float* __restrict__ Q, const float* __restrict__ T,
                            float* __restrict__ S4) {
  const int a  = blockIdx.x;        // 0..15 (uniform per block)
  const int bc = threadIdx.x;       // 0..255
  const int ab = a * 16 + (bc >> 4);
  const int cc = bc & 15;
  const int m  = a * 256 + bc;      // flat level-3 index (a,b,c)
  float u = 0.0f;
  float4 s0 = {0,0,0,0}, s1 = {0,0,0,0}, s2 = {0,0,0,0}, s3 = {0,0,0,0};
  for (int j = 0; j < NSTEP; ++j) {
    const float* dr = dx + j * DIMS;
    float4 d0 = *(const float4*)(dr + 0);
    float4 d1 = *(const float4*)(dr + 4);
    float4 d2 = *(const float4*)(dr + 8);
    float4 d3 = *(const float4*)(dr + 12);
    float qbc = Q[j * L2SZ + bc];
    float tab = T[j * L2SZ + ab];
    float ca  = cx[j * DIMS + a];
    float dxa = dr[a];
    float dxc = dr[cc];
    s0.x += u * d0.x; s0.y += u * d0.y; s0.z += u * d0.z; s0.w += u * d0.w;
    s1.x += u * d1.x; s1.y += u * d1.y; s1.z += u * d1.z; s1.w += u * d1.w;
    s2.x += u * d2.x; s2.y += u * d2.y; s2.z += u * d2.z; s2.w += u * d2.w;
    s3.x += u * d3.x; s3.y += u * d3.y; s3.z += u * d3.z; s3.w += u * d3.w;
    u += tab * dxc + (ca + dxa * (1.0f / 3.0f)) * qbc;
  }
  float4* o = (float4*)(S4 + m * 16);
  o[0] = s0; o[1] = s1; o[2] = s2; o[3] = s3;
}

// Stage a KCH x 16 operand tile (row-major, leading dim ld) into LDS.
// 4 lanes per row, 8 rows per b128 transfer, 4 transfers per tile.
__device__ __forceinline__ void stage_tile(float* dst /*LDS*/,
                                           const float* __restrict__ src,
                                           int ld, int lane) {
  const int srow = lane >> 2;        // 0..7
  const int scol = (lane & 3) << 2;  // 0,4,8,12
#pragma unroll
  for (int i = 0; i < 4; ++i) {
    const int row = srow + i * 8;
    const float* g = src + row * ld + scol;
    float* l = dst + row * 16 + scol;
#if USE_ASYNC_LDS
    __builtin_amdgcn_global_load_async_to_lds_b128(
        (__attribute__((address_space(1))) v4i_vs*)const_cast<float*>(g),
        (__attribute__((address_space(3))) v4i_vs*)l, 0, 0);
#else
    *(float4*)l = *(const float4*)g;
#endif
  }
}

// Out(M×N, ldo) += alpha · Aᵀ·B over K4 steps. One wave32 per 16×16 tile,
// K reduced in KCH chunks, double-buffered through LDS (async copy if available).
// MODE==1: B is Q (ldb=256) scaled on the fly by dx[k][aa]/3 — the virtual
// level-3 exponential term R[k][(a,bc)] = dx[k][a]·Q[k][bc]/3.
template <int MODE>
__global__ void k_gemm_atb(const float* __restrict__ A, int lda,
                           const float* __restrict__ B, int ldb,
                           float* __restrict__ out, int ldo, float alpha,
                           const float* __restrict__ dxp) {
  alignas(16) __shared__ float lA[2][KCH * 16];
  alignas(16) __shared__ float lB[2][KCH * 16];
  const int lane = threadIdx.x;          // blockDim = 32, uniform flow -> EXEC all-1s
  const int l15  = lane & 15;
  const int koff = (lane >> 4) << 1;     // lanes 16..31 hold K+2, K+3 (A/B operand layout)
  const int colA = blockIdx.y * 16;
  const int nB0  = blockIdx.x * 16;
  const int colB = (MODE == 0) ? nB0 : (nB0 & 255);
  const int aa   = nB0 >> 8;             // MODE==1 only

  stage_tile(lA[0], A + colA, lda, lane);
  stage_tile(lB[0], B + colB, ldb, lane);

  v8f acc = (v8f){0, 0, 0, 0, 0, 0, 0, 0};
  for (int kc = 0; kc < K4; kc += KCH) {
    const int cur = (kc / KCH) & 1;
    if (kc + KCH < K4) {
      stage_tile(lA[cur ^ 1], A + (kc + KCH) * lda + colA, lda, lane);
      stage_tile(lB[cur ^ 1], B + (kc + KCH) * ldb + colB, ldb, lane);
#if USE_ASYNC_LDS
      __builtin_amdgcn_s_wait_asynccnt(8);   // current chunk done; prefetch in flight
#endif
    } else {
#if USE_ASYNC_LDS
      __builtin_amdgcn_s_wait_asynccnt(0);
#endif
    }
#pragma unroll
    for (int kk = 0; kk < KCH; kk += 4) {
      v2f av, bv;
      av.x = lA[cur][(kk + koff    ) * 16 + l15];
      av.y = lA[cur][(kk + koff + 1) * 16 + l15];
      bv.x = lB[cur][(kk + koff    ) * 16 + l15];
      bv.y = lB[cur][(kk + koff + 1) * 16 + l15];
      if constexpr (MODE == 1) {
        bv.x *= dxp[(kc + kk + koff    ) * DIMS + aa] * (1.0f / 3.0f);
        bv.y *= dxp[(kc + kk + koff + 1) * DIMS + aa] * (1.0f / 3.0f);
      }
      acc = __builtin_amdgcn_wmma_f32_16x16x4_f32(
          /*neg_a=*/false, av, /*neg_b=*/false, bv,
          /*c_mod=*/(short)0, acc, /*reuse_a=*/false, /*reuse_b=*/false);
    }
  }
  const int mhi = (lane >> 4) << 3;      // C/D layout: lanes 16..31 -> M = r + 8
#pragma unroll
  for (int r = 0; r < 8; ++r) {
    int mm = blockIdx.y * 16 + mhi + r;
    out[mm * ldo + nB0 + l15] += alpha * acc[r];
  }
}

extern "C" void kernel_launch(void* const* d_in, const int* in_sizes, int n_in,
                              void* d_out, int out_size, void* d_ws, size_t ws_size,
                              hipStream_t stream) {
  (void)in_sizes; (void)n_in; (void)ws_size;
  const float* x = (const float*)d_in[0];
  float* out = (float*)d_out;
  float* ws  = (float*)d_ws;

  float* dx = ws;                    // K4*16
  float* cx = dx + K4 * DIMS;        // K4*16
  float* Q  = cx + K4 * DIMS;        // K4*256
  float* T  = Q  + K4 * L2SZ;        // K4*256   (total ≈ 8.9 MB workspace)

  float* S2 = out + 16;
  float* S3 = out + 16 + 256;
  float* S4 = out + 16 + 256 + 4096;

  k_zero<<<(out_size + 255) / 256, 256, 0, stream>>>(out, out_size);
  k_dx_q<<<K4, 256, 0, stream>>>(x, dx, Q);
  k_scan_c<<<1, 32, 0, stream>>>(dx, cx, out);
  k_scan_t<<<1, 256, 0, stream>>>(dx, cx, Q, T);
  k_scan_u_s4<<<16, 256, 0, stream>>>(dx, cx, Q, T, S4);   // S4 term 1 (writes '=')

  // S2 = CᵀD + 0.5·DᵀD                         (16×16)
  k_gemm_atb<0><<<dim3(1, 1),  32, 0, stream>>>(cx, DIMS, dx, DIMS, S2, DIMS, 1.0f,      nullptr);
  k_gemm_atb<0><<<dim3(1, 1),  32, 0, stream>>>(dx, DIMS, dx, DIMS, S2, DIMS, 0.5f,      nullptr);
  // S3 = TᵀD (256×16) + CᵀQ (16×256) + ⅓·DᵀQ (16×256)
  k_gemm_atb<0><<<dim3(1, 16), 32, 0, stream>>>(T,  L2SZ, dx, DIMS, S3, DIMS, 1.0f,      nullptr);
  k_gemm_atb<0><<<dim3(16, 1), 32, 0, stream>>>(cx, DIMS, Q,  L2SZ, S3, L2SZ, 1.0f,      nullptr);
  k_gemm_atb<0><<<dim3(16, 1), 32, 0, stream>>>(dx, DIMS, Q,  L2SZ, S3, L2SZ, 1.0f/3.0f, nullptr);
  // S4 += TᵀQ (256×256) + ⅙·QᵀQ (256×256) + CᵀR (16×4096, R generated on the fly)
  k_gemm_atb<0><<<dim3(16, 16), 32, 0, stream>>>(T,  L2SZ, Q,  L2SZ, S4, L2SZ, 1.0f,      nullptr);
  k_gemm_atb<0><<<dim3(16, 16), 32, 0, stream>>>(Q,  L2SZ, Q,  L2SZ, S4, L2SZ, 1.0f/6.0f, nullptr);
  k_gemm_atb<1><<<dim3(256, 1), 32, 0, stream>>>(cx, DIMS, Q,  L2SZ, S4, L3SZ, 1.0f,      dx);
}